// MVOnlyGATrBilinear_82678120448614
// MI455X (gfx1250) — compile-verified
//
#include <hip/hip_runtime.h>

typedef __attribute__((ext_vector_type(2))) float v2f;
typedef __attribute__((ext_vector_type(4))) float v4f;
typedef __attribute__((ext_vector_type(8))) float v8f;
typedef __attribute__((ext_vector_type(4))) unsigned int u32x4;
typedef __attribute__((ext_vector_type(8))) int i32x8;
typedef __attribute__((ext_vector_type(4))) int i32x4;

// ---------------------------------------------------------------------------
// Compile-time PGA G(3,0,1) blade algebra (blade order matches the reference).
// ---------------------------------------------------------------------------
constexpr int MASK_OF[16] = {0,1,2,4,8,3,5,9,6,10,12,7,11,13,14,15};
constexpr int IDX_OF[16]  = {0,1,2,5,3,6,8,11,4,7,9,12,10,13,14,15};

constexpr int cpop(int x) { int c = 0; for (int i = 0; i < 4; ++i) c += (x >> i) & 1; return c; }
constexpr int cross_inv(int a, int b) {
  int s = 0;
  for (int x = 0; x < 4; ++x) if (a & (1 << x)) s += cpop(b & ((1 << x) - 1));
  return s;
}
constexpr int gp_sign(int i, int j) {
  const int a = MASK_OF[i], b = MASK_OF[j];
  if (a & b & 1) return 0;
  int s = 0;
  for (int t = 0; t < 4; ++t) if (b & (1 << t)) s += cpop(a >> (t + 1));
  return (s & 1) ? -1 : 1;
}
constexpr int gp_idx(int i, int j) { return IDX_OF[MASK_OF[i] ^ MASK_OF[j]]; }
constexpr int dsign(int m) { return (cross_inv(m, (~m) & 15) & 1) ? -1 : 1; }
constexpr int jc_sign(int i, int j) {
  const int a = MASK_OF[i], b = MASK_OF[j];
  const int da = (~a) & 15, db = (~b) & 15;
  if (da & db) return 0;
  const int w = (cross_inv(da, db) & 1) ? -1 : 1;
  return dsign(a) * dsign(b) * w * dsign(a & b);
}
constexpr int jc_idx(int i, int j) { return IDX_OF[MASK_OF[i] & MASK_OF[j]]; }

// LDS geometry (dwords)
#define XTP   516     // transposed x tile: per-token pitch (16 comp * 32 ch + 4)
#define OTP   512     // output staging pitch (flat, TDM-stored)
#define HTP   1028    // h tile: per-token pitch (16 comp * 64 ch + 4)
#define WPT   34      // weight row pitch (32 ch + 2)
#define WBT_DW  (9 * 64 * WPT)              // 19584
#define WOT_DW  (9 * 32 * WPT)              // 9792
#define WAVE_DW (16 * XTP + 16 * HTP)       // 24704
#define SMEM_DW (WBT_DW + WOT_DW + 2 * WAVE_DW)  // 78784 dw = 315,136 B

// TDM D# group 0: count=1 user descriptor, LDS byte addr, 57-bit global addr, type=2
static __device__ inline u32x4 tdm_g0(unsigned ldsByte, const void* gptr) {
  const unsigned long long ga = (unsigned long long)(uintptr_t)gptr;
  u32x4 g0;
  g0.x = 1u;
  g0.y = ldsByte;
  g0.z = (unsigned)ga;
  g0.w = (unsigned)((ga >> 32) & 0x01FFFFFFull) | (2u << 30);
  return g0;
}

// ---------------------------------------------------------------------------
// Two waves per workgroup, one 16-token tile per wave.
//  - x tile arrives via ONE tensor_load_to_lds: 3-D gather tile
//    (X=1 elem, Y=32 channels @ stride 16, Z=16 comps @ stride 1) iterated
//    16x with lds_incr=516 dw / global_incr=512 dw -> transposed + bank-padded.
//  - output leaves via ONE tensor_store_from_lds (1-D 8192-dword tile).
//  - WMMA = V_WMMA_F32_16X16X4_F32; every operand is a contiguous b64 pair.
// ---------------------------------------------------------------------------
__global__ __launch_bounds__(64) void gatr_mvbilinear_kernel(
    const float* __restrict__ xg, const float* __restrict__ refg,
    const float* __restrict__ wb, const float* __restrict__ bb,
    const float* __restrict__ wo, const float* __restrict__ bo,
    float* __restrict__ outg)
{
  __shared__ float smem[SMEM_DW];
  const int tid  = threadIdx.x;
  const int wave = tid >> 5;
  const int lane = tid & 31;
  const int half = lane >> 4;
  const int lm   = lane & 15;

  float* wbT = smem;                            // [a][o][i], pitch WPT
  float* woT = smem + WBT_DW;
  const unsigned xtOffDw = (unsigned)(WBT_DW + WOT_DW) + (unsigned)wave * WAVE_DW;
  float* xT  = smem + xtOffDw;                  // [t][comp][ch32] (also out staging)
  float* hT  = xT + 16 * XTP;                   // [t][comp][ch64]

  const long tileBase = ((long)blockIdx.x * 2 + wave) * 16;
  const i32x8 gPad = {0, 0, 0, 0, 0, 0, 0, 0}; // 5th builtin arg (zero-filled)

  // ---- kick off the async transposing x-tile load on the Tensor Data Mover --
  {
    u32x4 g0 = tdm_g0(xtOffDw * 4u, xg + tileBase * 512);
    i32x8 g1 = { (2 << 16) | (1 << 19),     // data_size=4B, iterate_enable
                 (int)0xFFFF0000u,          // barrier=0; tensor_dim0 = 0xFFFF
                 (int)0xFFFF0000u,          // dim0 hi=0; tensor_dim1 = 0xFFFF
                 0x00010000,                // dim1 hi=0; tile_dim0 = 1
                 32 | (16 << 16),           // tile_dim1 = 32 ch, tile_dim2 = 16 comp
                 16,                        // tensor_dim0_stride = 16 (comp pitch)
                 1 << 16,                   // dim0_str hi=0; tensor_dim1_stride = 1
                 0 };
    i32x4 g2 = { 0xFFFF,                    // tensor_dim2
                 516,                       // lds_addr_increment per token (dw)
                 512,                       // global_addr_increment per token (dw)
                 15 << 16 };                // iterate_count = 15 -> 16 tokens
    i32x4 g3 = { 0, 0, 0, 0 };
    __builtin_amdgcn_tensor_load_to_lds(g0, g1, g2, g3, gPad, 0);
  }

  // ---- cooperative weight transpose into LDS (overlaps the TDM load) ----
  for (int it = tid; it < (64 * 32 * 9) / 4; it += 64) {
    v4f v = *(const v4f*)(wb + it * 4);
    #pragma unroll
    for (int q = 0; q < 4; ++q) {
      const int flat = it * 4 + q;        // (o*32 + i)*9 + a
      const int a  = flat % 9;
      const int oi = flat / 9;
      wbT[(a * 64 + (oi >> 5)) * WPT + (oi & 31)] = v[q];
    }
  }
  for (int it = tid; it < (32 * 32 * 9) / 4; it += 64) {
    v4f v = *(const v4f*)(wo + it * 4);
    #pragma unroll
    for (int q = 0; q < 4; ++q) {
      const int flat = it * 4 + q;
      const int a  = flat % 9;
      const int oi = flat / 9;
      woT[(a * 32 + (oi >> 5)) * WPT + (oi & 31)] = v[q];
    }
  }
  __syncthreads();                         // weights visible to both waves
  __builtin_amdgcn_s_wait_tensorcnt(0);    // x tile resident in LDS

  // =======================  Layer 1: equi_linear(x, w_bil)  =================
  #pragma unroll
  for (int k = 0; k < 16; ++k) {
    const int  m    = MASK_OF[k];
    const int  g    = cpop(m);
    const bool e0   = (m & 1) != 0;
    const int  inC2 = IDX_OF[m ^ 1];
    v2f Ag[8], Ae[8];
    #pragma unroll
    for (int s = 0; s < 8; ++s) {
      Ag[s] = *(const v2f*)&xT[lm * XTP + k * 32 + s * 4 + half * 2];
      if (e0)
        Ae[s] = *(const v2f*)&xT[lm * XTP + inC2 * 32 + s * 4 + half * 2];
    }
    for (int nt = 0; nt < 4; ++nt) {
      v8f acc = {0.f,0.f,0.f,0.f,0.f,0.f,0.f,0.f};
      const int o = nt * 16 + lm;
      #pragma unroll
      for (int s = 0; s < 8; ++s) {
        v2f B = *(const v2f*)&wbT[(g * 64 + o) * WPT + s * 4 + half * 2];
        acc = __builtin_amdgcn_wmma_f32_16x16x4_f32(
            false, Ag[s], false, B, (short)0, acc, false, false);
      }
      if (e0) {
        #pragma unroll
        for (int s = 0; s < 8; ++s) {
          v2f B = *(const v2f*)&wbT[((g + 4) * 64 + o) * WPT + s * 4 + half * 2];
          acc = __builtin_amdgcn_wmma_f32_16x16x4_f32(
              false, Ae[s], false, B, (short)0, acc, false, false);
        }
      }
      if (k == 0) {
        const float bias = bb[o];
        #pragma unroll
        for (int r = 0; r < 8; ++r) acc[r] += bias;
      }
      #pragma unroll
      for (int r = 0; r < 8; ++r)
        hT[(half * 8 + r) * HTP + k * 64 + o] = acc[r];
    }
  }

  // ==========  Bilinear: geometric product + (ref-scaled) join  =============
  const float refv = refg[(tileBase + lm) * 16 + 15];
  for (int p = 0; p < 8; ++p) {
    const int c = half * 8 + p;
    const int base = lm * HTP + c;
    float lg[16], rg[16], lj[16], rj[16];
    #pragma unroll
    for (int i = 0; i < 16; ++i) {
      lg[i] = hT[base + i * 64];
      rg[i] = hT[base + i * 64 + 16];
      lj[i] = hT[base + i * 64 + 32];
      rj[i] = hT[base + i * 64 + 48];
    }
    float gp[16], jn[16];
    #pragma unroll
    for (int k = 0; k < 16; ++k) { gp[k] = 0.0f; jn[k] = 0.0f; }
    #pragma unroll
    for (int i = 0; i < 16; ++i) {
      #pragma unroll
      for (int j = 0; j < 16; ++j) {
        {
          const int s = gp_sign(i, j);
          if (s) {
            const int k = gp_idx(i, j);
            gp[k] = (s > 0) ? fmaf(lg[i], rg[j], gp[k]) : fmaf(-lg[i], rg[j], gp[k]);
          }
        }
        {
          const int s = jc_sign(i, j);
          if (s) {
            const int k = jc_idx(i, j);
            jn[k] = (s > 0) ? fmaf(lj[i], rj[j], jn[k]) : fmaf(-lj[i], rj[j], jn[k]);
          }
        }
      }
    }
    #pragma unroll
    for (int i = 0; i < 16; ++i) {
      hT[base + i * 64]      = gp[i];
      hT[base + i * 64 + 32] = jn[i] * refv;
    }
  }

  // ===============  Layer 2: equi_linear(concat[gp, join], w_out)  ==========
  #pragma unroll
  for (int k = 0; k < 16; ++k) {
    const int  m    = MASK_OF[k];
    const int  g    = cpop(m);
    const bool e0   = (m & 1) != 0;
    const int  inC2 = IDX_OF[m ^ 1];
    v2f Ag[8], Ae[8];
    #pragma unroll
    for (int s = 0; s < 8; ++s) {
      const int craw = s * 4 + half * 2;
      const int cc   = (s < 4) ? craw : craw + 16;
      Ag[s] = *(const v2f*)&hT[lm * HTP + k * 64 + cc];
      if (e0)
        Ae[s] = *(const v2f*)&hT[lm * HTP + inC2 * 64 + cc];
    }
    for (int nt = 0; nt < 2; ++nt) {
      v8f acc = {0.f,0.f,0.f,0.f,0.f,0.f,0.f,0.f};
      const int o = nt * 16 + lm;
      #pragma unroll
      for (int s = 0; s < 8; ++s) {
        v2f B = *(const v2f*)&woT[(g * 32 + o) * WPT + s * 4 + half * 2];
        acc = __builtin_amdgcn_wmma_f32_16x16x4_f32(
            false, Ag[s], false, B, (short)0, acc, false, false);
      }
      if (e0) {
        #pragma unroll
        for (int s = 0; s < 8; ++s) {
          v2f B = *(const v2f*)&woT[((g + 4) * 32 + o) * WPT + s * 4 + half * 2];
          acc = __builtin_amdgcn_wmma_f32_16x16x4_f32(
              false, Ae[s], false, B, (short)0, acc, false, false);
        }
      }
      if (k == 0) {
        const float bias = bo[o];
        #pragma unroll
        for (int r = 0; r < 8; ++r) acc[r] += bias;
      }
      #pragma unroll
      for (int r = 0; r < 8; ++r)      // stage output tile [t][ch][comp], flat
        xT[(half * 8 + r) * OTP + o * 16 + k] = acc[r];
    }
  }

  // ---- TDM store of the staged 16x32x16 tile (one 8192-dword line) ----
  asm volatile("s_wait_dscnt 0x0" ::: "memory");   // staging writes complete
  {
    u32x4 g0 = tdm_g0(xtOffDw * 4u, outg + tileBase * 512);
    i32x8 g1 = { (2 << 16),                // data_size=4B, no iterate
                 (int)0xFFFF0000u,         // tensor_dim0 = 0xFFFF
                 (int)0xFFFF0000u,         // tensor_dim1 = 0xFFFF
                 (int)(8192 << 16),        // tile_dim0 = 8192
                 0,                        // tile_dim1/2 unused (1-D)
                 8192,                     // tensor_dim0_stride
                 0, 0 };
    i32x4 g2 = { 0xFFFF, 0, 0, 0 };
    i32x4 g3 = { 0, 0, 0, 0 };
    __builtin_amdgcn_tensor_store_from_lds(g0, g1, g2, g3, gPad, 0);
  }
  // S_ENDPGM performs an implicit S_WAIT_IDLE, covering the outstanding TDM op.
}

extern "C" void kernel_launch(void* const* d_in, const int* in_sizes, int n_in,
                              void* d_out, int out_size, void* d_ws, size_t ws_size,
                              hipStream_t stream) {
  (void)n_in; (void)out_size; (void)d_ws; (void)ws_size;
  const float* x   = (const float*)d_in[0];
  const float* ref = (const float*)d_in[1];
  const float* wb  = (const float*)d_in[2];
  const float* bb  = (const float*)d_in[3];
  const float* wo  = (const float*)d_in[4];
  const float* bo  = (const float*)d_in[5];
  float* out = (float*)d_out;

  const int tokens = in_sizes[0] / 512;   // x is [tokens, 32, 16]
  const int blocks = tokens / 32;         // 2 waves * 16 tokens per workgroup
  hipLaunchKernelGGL(gatr_mvbilinear_kernel, dim3(blocks), dim3(64), 0, stream,
                     x, ref, wb, bb, wo, bo, out);
}